// Model_62809601737090
// MI455X (gfx1250) — compile-verified
//
#include <hip/hip_runtime.h>
#include <math.h>

typedef float v2f __attribute__((ext_vector_type(2)));
typedef float v8f __attribute__((ext_vector_type(8)));
typedef unsigned int v4u __attribute__((ext_vector_type(4)));
typedef int v8i __attribute__((ext_vector_type(8)));
typedef int v4i __attribute__((ext_vector_type(4)));

#define Bn 128
#define Cn 3
#define Hn 256
#define Wn 256
#define Dn 3072          // C * 32 * 32 per grid cell
#define Pn 32
#define CHUNK 64         // K-chunk (2 image rows per cell)
#define NCH (Dn / CHUNK) // 48
#define AROW 68          // 64 dwords + 4 pad -> 16B aligned, conflict-free
#define ABUF (32 * AROW) // floats per A buffer (32 cells)
#define BBUF (32 * AROW) // floats per B buffer (32 comps)

#if __has_builtin(__builtin_amdgcn_tensor_load_to_lds)
#define USE_TDM 1
#else
#define USE_TDM 0
#endif

// ---------------------------------------------------------------------------
// Kernel 0: mean_proj[p] = sum_d pca_mean[d] * comp[p][d]
// ---------------------------------------------------------------------------
__global__ void k_meanproj(const float* __restrict__ mean,
                           const float* __restrict__ comp,
                           float* __restrict__ mp) {
    __shared__ float red[128];
    int p = blockIdx.x;
    int t = threadIdx.x;
    float s = 0.f;
    for (int d = t; d < Dn; d += 128) s += mean[d] * comp[p * Dn + d];
    red[t] = s;
    __syncthreads();
    for (int off = 64; off > 0; off >>= 1) {
        if (t < off) red[t] += red[t + off];
        __syncthreads();
    }
    if (t == 0) mp[p] = red[0];
}

// ---------------------------------------------------------------------------
// Kernel 1: per-(cell,k) Cholesky + triangular inverse + half log-det.
// One wave32 per 32x32 matrix; 4 waves (128 threads) per block, 32 blocks.
// ---------------------------------------------------------------------------
__global__ void __launch_bounds__(128) k_chol(const float* __restrict__ cov,
                                              float* __restrict__ linv,
                                              float* __restrict__ ldet) {
    __shared__ float Lm[4][32][33];
    __shared__ float Li[4][32][33];
    int wid  = threadIdx.x >> 5;
    int lane = threadIdx.x & 31;
    int idx  = blockIdx.x * 4 + wid;       // 0..127 = cell*2 + k
    float (*L)[33] = Lm[wid];
    float (*X)[33] = Li[wid];

    for (int c = 0; c < 32; ++c)
        L[lane][c] = cov[idx * 1024 + lane * 32 + c];
    __builtin_amdgcn_wave_barrier();

    for (int k = 0; k < 32; ++k) {
        float dk = sqrtf(L[k][k]);
        if (lane == k)      L[k][k] = dk;
        else if (lane > k)  L[lane][k] = L[lane][k] / dk;
        __builtin_amdgcn_wave_barrier();
        float lrk = (lane > k) ? L[lane][k] : 0.f;
        for (int c = k + 1; c <= lane; ++c)
            L[lane][c] -= lrk * L[c][k];
        __builtin_amdgcn_wave_barrier();
    }

    float ld = logf(L[lane][lane]);
    for (int m = 16; m >= 1; m >>= 1) ld += __shfl_xor(ld, m, 32);
    if (lane == 0) ldet[idx] = ld;

    int j = lane;                           // lane = column of L^-1
    for (int r = 0; r < j; ++r) X[r][j] = 0.f;
    X[j][j] = 1.0f / L[j][j];
    for (int r = j + 1; r < 32; ++r) {
        float s = 0.f;
        for (int t2 = j; t2 < r; ++t2) s += L[r][t2] * X[t2][j];
        X[r][j] = -s / L[r][r];
    }
    __builtin_amdgcn_wave_barrier();
    for (int c = 0; c < 32; ++c)
        linv[idx * 1024 + lane * 32 + c] = X[lane][c];
}

// ---------------------------------------------------------------------------
// TDM helper: one tensor_load_to_lds with a 4D (or 2D) tile descriptor.
// data_size=4B, pad_enable=1 (padIntv code, padAmt code), count=1, type=2.
// strides in data_size units; tile LDS-packed X->Y->Z->ZZ.
// clang-23 builtin arity: (v4u g0, v8i g1, v4i g2, v4i g3, v8i extra, i32 cpol)
// ---------------------------------------------------------------------------
#if USE_TDM
__device__ __forceinline__ void tdm_load_tile(unsigned ldsByteOff,
                                              const void* gptr,
                                              unsigned tile0, unsigned tile1,
                                              unsigned tile2, unsigned tile3,
                                              unsigned long long s0,
                                              unsigned long long s1,
                                              unsigned long long s2,
                                              unsigned padIntv, unsigned padAmt) {
    unsigned long long ga = (unsigned long long)(size_t)gptr;
    const unsigned TD = 1u << 30;          // huge tensor dims: no OOB zero-fill
    v4u g0;
    g0[0] = 1u;                                        // count=1, user mode
    g0[1] = ldsByteOff;                                // lds_addr
    g0[2] = (unsigned)ga;                              // global_addr[31:0]
    g0[3] = (unsigned)((ga >> 32) & 0x01FFFFFFu) | (2u << 30); // [56:32]|type=2
    v8i g1;
    g1[0] = (int)((2u << 16) | (1u << 20) | (padIntv << 22) | (padAmt << 25));
    g1[1] = (int)((TD & 0xFFFFu) << 16);               // tensor_dim0[15:0]
    g1[2] = (int)(((TD >> 16) & 0xFFFFu) | ((TD & 0xFFFFu) << 16));
    g1[3] = (int)(((TD >> 16) & 0xFFFFu) | (tile0 << 16));
    g1[4] = (int)(tile1 | (tile2 << 16));
    g1[5] = (int)(s0 & 0xFFFFFFFFull);                 // dim0 stride lo
    g1[6] = (int)(((s0 >> 32) & 0xFFFFull) | ((s1 & 0xFFFFull) << 16));
    g1[7] = (int)((s1 >> 16) & 0xFFFFFFFFull);
    v4i g2;
    g2[0] = (int)TD;                                   // tensor_dim2
    g2[1] = (int)TD;                                   // tensor_dim3
    g2[2] = (int)(s2 & 0xFFFFFFFFull);                 // dim2 stride lo
    g2[3] = (int)(((s2 >> 32) & 0xFFFFull) | (tile3 << 16));
    v4i g3 = {0, 0, 0, 0};                             // no 5th dim
    v8i g4 = {0, 0, 0, 0, 0, 0, 0, 0};                 // toolchain extra group
    __builtin_amdgcn_tensor_load_to_lds(g0, g1, g2, g3, g4, 0);
}
#endif

// ---------------------------------------------------------------------------
// Kernel 2: PCA GEMM via V_WMMA_F32_16X16X4_F32 with TDM-fed double-buffered
// LDS. One block per (image b, 32-cell half); 128 threads = 4 waves:
// wave = (mtile = wid&1, nhalf = wid>>1). 48 chunks x 16 WMMA = 768 per wave.
// ---------------------------------------------------------------------------
__global__ void __launch_bounds__(128) k_pca(const float* __restrict__ img,
                                             const float* __restrict__ comp,
                                             const float* __restrict__ mp,
                                             float* __restrict__ z) {
    __shared__ float lds[2 * ABUF + 2 * BBUF];   // 34.8 KB

    int t    = threadIdx.x;
    int wid  = t >> 5;
    int lane = t & 31;
    int b    = blockIdx.x >> 1;
    int half = blockIdx.x & 1;                   // cells half*32 .. +31
    int mtile = wid & 1;
    int nhalf = wid >> 1;
    int lr = lane & 15;
    int lh = lane >> 4;

    v8f acc = {0.f, 0.f, 0.f, 0.f, 0.f, 0.f, 0.f, 0.f};
    const float* imgb = img + (size_t)b * (Cn * Hn * Wn) + half * 128 * Wn;

    auto stage = [&](int kc) {
        int c  = kc >> 4;                 // channel (64 | 1024)
        int i0 = (kc & 15) * 2;           // 2 rows per cell per chunk
        int cur = kc & 1;
#if USE_TDM
        if (wid == 0) {
            // Real pointer-derived LDS offsets: captures the LDS array so the
            // compiler keeps the ds_loads (TDM writes are invisible to LLVM).
            unsigned aoff = (unsigned)(size_t)(void*)&lds[cur * ABUF];
            unsigned boff = (unsigned)(size_t)(void*)&lds[2 * ABUF + cur * BBUF];
            // A: 4D tile (32 j, 2 rows, 8 gx, 4 gy), strides (256, 32, 8192)
            tdm_load_tile(aoff, imgb + c * (Hn * Wn) + i0 * Wn,
                          32, 2, 8, 4, 256ull, 32ull, 8192ull, 5, 3);
            // B: 2D tile (64 k, 32 p), stride 3072
            tdm_load_tile(boff, comp + kc * CHUNK,
                          64, 32, 0, 0, 3072ull, 0ull, 0ull, 5, 3);
        }
        asm volatile("" ::: "memory");    // LDS modified by TDM engine
#else
        float* Ab = &lds[cur * ABUF];
        for (int it = 0; it < 4; ++it) {
            int idx  = it * 128 + t;      // 0..511 float4
            int cell = idx >> 4;
            int q    = idx & 15;
            int rowi = q >> 3, j4 = q & 7;
            int gy = cell >> 3, gx = cell & 7;
            const float* src = imgb + c * (Hn * Wn)
                             + (gy * 32 + i0 + rowi) * Wn + gx * 32 + j4 * 4;
            *(float4*)&Ab[cell * AROW + rowi * 32 + j4 * 4] = *(const float4*)src;
        }
        float* Bb = &lds[2 * ABUF + cur * BBUF];
        for (int it = 0; it < 4; ++it) {
            int idx = it * 128 + t;
            int p = idx >> 4, q = idx & 15;
            *(float4*)&Bb[p * AROW + q * 4] =
                *(const float4*)(comp + p * Dn + kc * CHUNK + q * 4);
        }
#endif
    };

    auto compute = [&](int kc) {
        int cur = kc & 1;
        const float* Ab = &lds[cur * ABUF + (mtile * 16 + lr) * AROW];
        const float* Bb = &lds[2 * ABUF + cur * BBUF + (nhalf * 16 + lr) * AROW];
        #pragma unroll
        for (int kk = 0; kk < CHUNK; kk += 4) {
            int ka = kk + 2 * lh;         // 16x4 f32 A layout: K pair per lane
            v2f av = *(const v2f*)&Ab[ka];
            v2f bv = *(const v2f*)&Bb[ka];
            acc = __builtin_amdgcn_wmma_f32_16x16x4_f32(
                false, av, false, bv, (short)0, acc, false, false);
        }
    };

    stage(0);
    for (int kc = 0; kc < NCH - 1; ++kc) {
        __syncthreads();                  // buffer kc+1 free: all readers done
        stage(kc + 1);                    // prefetch next chunk
#if USE_TDM
        __builtin_amdgcn_s_wait_tensorcnt((short)2);  // chunk kc landed
        asm volatile("" ::: "memory");
#endif
        __syncthreads();
        compute(kc);
    }
#if USE_TDM
    __builtin_amdgcn_s_wait_tensorcnt((short)0);      // drain final chunk
    asm volatile("" ::: "memory");
#endif
    __syncthreads();
    compute(NCH - 1);

    // C/D layout: VGPR v -> row v + 8*lh, col = lane&15 (+ nhalf*16)
    int n = nhalf * 16 + lr;
    float m = mp[n];
    int mbase = b * 64 + half * 32 + mtile * 16 + lh * 8;
    #pragma unroll
    for (int v = 0; v < 8; ++v)
        z[(mbase + v) * 32 + n] = acc[v] - m;
}

// ---------------------------------------------------------------------------
// Kernel 3: GMM scoring. One block per b; wave handles cells wid, wid+8, ...
// ---------------------------------------------------------------------------
__global__ void __launch_bounds__(256) k_score(const float* __restrict__ z,
                                               const float* __restrict__ linv,
                                               const float* __restrict__ ldet,
                                               const float* __restrict__ gw,
                                               const float* __restrict__ gmean,
                                               float* __restrict__ raw) {
    __shared__ float smax[8];
    int b    = blockIdx.x;
    int wid  = threadIdx.x >> 5;
    int lane = threadIdx.x & 31;
    const float C0 = 58.8120661435f;   // P * log(2*pi)
    float wmax = 0.0f;                 // clamp-at-0 folded into init

    for (int cell = wid; cell < 64; cell += 8) {
        int m = b * 64 + cell;
        float lp0 = 0.f, lp1 = 0.f;
        #pragma unroll
        for (int kg = 0; kg < 2; ++kg) {
            int idx = cell * 2 + kg;
            const float* Lr = linv + idx * 1024 + lane * 32;
            const float* zr = z + m * 32;            // wave-uniform
            const float* mu = gmean + idx * 32;      // wave-uniform
            float s = 0.f;
            #pragma unroll
            for (int cc = 0; cc < 32; ++cc)
                s += Lr[cc] * (zr[cc] - mu[cc]);
            float q = s * s;
            for (int msk = 16; msk >= 1; msk >>= 1) q += __shfl_xor(q, msk, 32);
            float lp = -0.5f * (C0 + q) - ldet[idx] + logf(gw[idx]);
            if (kg == 0) lp0 = lp; else lp1 = lp;
        }
        float a  = fmaxf(lp0, lp1);
        float ll = a + logf(expf(lp0 - a) + expf(lp1 - a));
        wmax = fmaxf(wmax, -ll);
    }
    if (lane == 0) smax[wid] = wmax;
    __syncthreads();
    if (threadIdx.x == 0) {
        float mx = smax[0];
        for (int i = 1; i < 8; ++i) mx = fmaxf(mx, smax[i]);
        raw[b] = mx;
    }
}

// ---------------------------------------------------------------------------
// Kernel 4: batch min-max normalize (B = 128 fits one block).
// ---------------------------------------------------------------------------
__global__ void k_norm(const float* __restrict__ raw, float* __restrict__ out) {
    __shared__ float mn[128], mx[128];
    int t = threadIdx.x;
    float v = raw[t];
    mn[t] = v; mx[t] = v;
    __syncthreads();
    for (int off = 64; off > 0; off >>= 1) {
        if (t < off) {
            mn[t] = fminf(mn[t], mn[t + off]);
            mx[t] = fmaxf(mx[t], mx[t + off]);
        }
        __syncthreads();
    }
    out[t] = (v - mn[0]) / (mx[0] - mn[0] + 1e-6f);
}

// ---------------------------------------------------------------------------
extern "C" void kernel_launch(void* const* d_in, const int* in_sizes, int n_in,
                              void* d_out, int out_size, void* d_ws, size_t ws_size,
                              hipStream_t stream) {
    const float* img   = (const float*)d_in[0];   // [128,3,256,256]
    const float* pmean = (const float*)d_in[1];   // [3072]
    const float* comp  = (const float*)d_in[2];   // [32,3072]
    const float* gw    = (const float*)d_in[3];   // [64,2]
    const float* gmean = (const float*)d_in[4];   // [64,2,32]
    const float* gcov  = (const float*)d_in[5];   // [64,2,32,32]
    float* out = (float*)d_out;                   // [128]

    float* wsf  = (float*)d_ws;
    float* mp   = wsf;                  // 32
    float* ldet = wsf + 32;             // 128
    float* raw  = wsf + 160;            // 128
    float* linv = wsf + 512;            // 128*1024
    float* z    = wsf + 512 + 128*1024; // 8192*32

    k_meanproj<<<32, 128, 0, stream>>>(pmean, comp, mp);
    k_chol    <<<32, 128, 0, stream>>>(gcov, linv, ldet);
    k_pca     <<<Bn * 2, 128, 0, stream>>>(img, comp, mp, z);
    k_score   <<<Bn, 256, 0, stream>>>(z, linv, ldet, gw, gmean, raw);
    k_norm    <<<1, 128, 0, stream>>>(raw, out);
}